// QuantumLayer_47588237640416
// MI455X (gfx1250) — compile-verified
//
#include <hip/hip_runtime.h>
#include <hip/hip_bf16.h>

// Quantum circuit simulator for MI455X (gfx1250, wave32).
// One workgroup (256 threads = 8 wave32) per batch sample; the 1024-amplitude
// complex state lives entirely in LDS. RY over 4-wire groups is applied as a
// dense 16x16 Kronecker matrix using chained V_WMMA_F32_16X16X4_F32 ops
// (fp32 WMMA keeps complex64 fidelity; the problem is latency bound, not
// matrix-throughput bound). RZ / encoding phases use the hardware
// v_sin_f32/v_cos_f32 path (__sincosf) -- angles are ~N(0, sqrt(10)/2) rad,
// far inside the hw range-reduction window, so no Payne-Hanek slow path.

#define N_WIRES  10
#define DIM      1024
#define NBLOCKS  4
#define THREADS  256
#define SPT      (DIM / THREADS)   // 4 states per thread

typedef __attribute__((ext_vector_type(2))) float v2f;
typedef __attribute__((ext_vector_type(8))) float v8f;

// fast hw sincos: v_sin_f32 / v_cos_f32
__device__ __forceinline__ void fast_sincos(float a, float* s, float* c) {
  __sincosf(a, s, c);
}

// ---- CNOT-ring permutation: P[s] = g_(0,1)(g_(1,2)(...g_(8,9)(g_(9,0)(s))))
__device__ __forceinline__ int perm_index(int s) {
  int idx = s;
  // last pair (c=9, t=0): control bit at position 0, flip bit 9
  int bit = idx & 1;
  idx ^= bit << 9;
  // pairs (c, c+1) for c = 8..0: control at 9-c, target at 8-c
  #pragma unroll
  for (int c = 8; c >= 0; --c) {
    int bc = (idx >> (9 - c)) & 1;
    idx ^= bc << (8 - c);
  }
  return idx;
}

// ---- diagonal RZ: state *= exp(-0.5i * theta.z) ; a = 0.5*(theta.z)
__device__ __forceinline__ void rz_stage(float* re, float* im, const float* th, int tid) {
  #pragma unroll
  for (int r = 0; r < SPT; ++r) {
    int s = tid + r * THREADS;
    float a = 0.f;
    #pragma unroll
    for (int w = 0; w < N_WIRES; ++w) {
      float t = th[w];
      a += ((s >> (9 - w)) & 1) ? -t : t;    // zval = 1 - 2*bit
    }
    a *= 0.5f;
    float sa, ca;
    fast_sincos(a, &sa, &ca);
    float r0 = re[s], i0 = im[s];
    re[s] = fmaf(i0, sa, r0 * ca);           // (r + i*m)*(ca - i*sa)
    im[s] = fmaf(-r0, sa, i0 * ca);
  }
}

// ---- build 16x16 Kronecker matrix of RY(theta) for wires w0..w0+3
// (bit 3 of the group nibble <-> wire w0). One element per thread.
__device__ __forceinline__ void build_M(float* Msh, const float* th, int w0, int tid) {
  int j = tid >> 4, i = tid & 15;
  float v = 1.f;
  #pragma unroll
  for (int q = 0; q < 4; ++q) {
    float h = 0.5f * th[w0 + q];
    float sw, cw;
    fast_sincos(h, &sw, &cw);
    int jb = (j >> (3 - q)) & 1;
    int ib = (i >> (3 - q)) & 1;
    // U = [[c,-s],[s,c]]
    v *= (jb == ib) ? cw : (jb ? sw : -sw);
  }
  Msh[tid] = v;
}

// ---- low group (bits 3..0): Snew(64x16) = S @ M^T, one 16-row tile per wave
__device__ __forceinline__ void wmma_low(float* plane, const float* Msh, int tile, int lane) {
  const int half = lane >> 4;        // lane half selects K pair / M half
  const int l15  = lane & 15;
  v8f acc = {0.f, 0.f, 0.f, 0.f, 0.f, 0.f, 0.f, 0.f};
  const float* rowbase = plane + (tile * 16 + l15) * 16;
  #pragma unroll
  for (int k0 = 0; k0 < 16; k0 += 4) {
    v2f a, b;
    int ks = k0 + 2 * half;
    // A[m=l15][k]: lanes 0-15 hold K=k0,k0+1; lanes 16-31 hold K=k0+2,k0+3
    a.x = rowbase[ks];
    a.y = rowbase[ks + 1];
    // B[k][n] = M^T[k][n] = M[n][k]; v0: rows k0(+half), v1: rows k0+2(+half)
    b.x = Msh[l15 * 16 + k0 + half];
    b.y = Msh[l15 * 16 + k0 + 2 + half];
    acc = __builtin_amdgcn_wmma_f32_16x16x4_f32(false, a, false, b,
                                                (short)0, acc, false, false);
  }
  // D layout: VGPR v -> M = v + 8*half, N = l15
  #pragma unroll
  for (int v = 0; v < 8; ++v)
    plane[(tile * 16 + v + 8 * half) * 16 + l15] = acc[v];
}

// ---- mid group (bits 7..4): Snew_o(16x16) = M2 @ S_o, one slab o per wave
__device__ __forceinline__ void wmma_mid(float* plane, const float* Msh, int o, int lane) {
  const int half = lane >> 4;
  const int l15  = lane & 15;
  v8f acc = {0.f, 0.f, 0.f, 0.f, 0.f, 0.f, 0.f, 0.f};
  float* base = plane + o * 256;
  #pragma unroll
  for (int k0 = 0; k0 < 16; k0 += 4) {
    v2f a, b;
    int ks = k0 + 2 * half;
    a.x = Msh[l15 * 16 + ks];              // A[m][k] = M2[m][k]
    a.y = Msh[l15 * 16 + ks + 1];
    b.x = base[(k0 + half) * 16 + l15];    // B[k][n] = S_o[k][n]
    b.y = base[(k0 + 2 + half) * 16 + l15];
    acc = __builtin_amdgcn_wmma_f32_16x16x4_f32(false, a, false, b,
                                                (short)0, acc, false, false);
  }
  #pragma unroll
  for (int v = 0; v < 8; ++v)
    base[(v + 8 * half) * 16 + l15] = acc[v];
}

// ---- full RY_all over 10 wires: two 16x16 WMMA groups + 2 butterfly wires
__device__ __forceinline__ void ry_all(float* re, float* im, float* Msh,
                                       const float* th, int tid) {
  const int lane = tid & 31;
  const int wv   = tid >> 5;        // 0..7
  float* P = (wv >> 2) ? im : re;   // 4 waves on real plane, 4 on imag
  const int unit = wv & 3;          // tile (low) or slab (mid)

  __syncthreads();                  // previous stage's state writes visible
  build_M(Msh, th, /*w0=*/6, tid);  // wires 6..9 <-> bits 3..0
  __syncthreads();
  wmma_low(P, Msh, unit, lane);
  __syncthreads();
  build_M(Msh, th, /*w0=*/2, tid);  // wires 2..5 <-> bits 7..4
  __syncthreads();
  wmma_mid(P, Msh, unit, lane);
  __syncthreads();

  // wires 0 and 1 (bits 9, 8): classic butterflies, 512 pairs / stage
  #pragma unroll
  for (int w = 0; w < 2; ++w) {
    float h = 0.5f * th[w];
    float sw, cw;
    fast_sincos(h, &sw, &cw);
    const int right = 1 << (9 - w);
    #pragma unroll
    for (int r = 0; r < 2; ++r) {
      int p  = tid + r * THREADS;
      int k  = p & (right - 1);
      int i  = p >> (9 - w);
      int s0 = i * (right << 1) + k;
      int s1 = s0 + right;
      float r0 = re[s0], r1 = re[s1], i0 = im[s0], i1 = im[s1];
      re[s0] = fmaf(cw, r0, -sw * r1);
      im[s0] = fmaf(cw, i0, -sw * i1);
      re[s1] = fmaf(sw, r0, cw * r1);
      im[s1] = fmaf(sw, i0, cw * i1);
    }
    __syncthreads();
  }
}

__device__ __forceinline__ void apply_perm(float* re, float* im,
                                           const unsigned short* permsh, int tid) {
  float tr[SPT], ti[SPT];
  __syncthreads();
  #pragma unroll
  for (int r = 0; r < SPT; ++r) {
    int s  = tid + r * THREADS;
    int ps = permsh[s];
    tr[r] = re[ps];
    ti[r] = im[ps];
  }
  __syncthreads();
  #pragma unroll
  for (int r = 0; r < SPT; ++r) {
    int s = tid + r * THREADS;
    re[s] = tr[r];
    im[s] = ti[r];
  }
  __syncthreads();
}

__device__ __forceinline__ void apply_enc(float* re, float* im,
                                          const float* ec, const float* es, int tid) {
  #pragma unroll
  for (int r = 0; r < SPT; ++r) {
    int s = tid + r * THREADS;
    float ca = ec[s], sa = es[s];
    float r0 = re[s], i0 = im[s];
    re[s] = fmaf(i0, sa, r0 * ca);
    im[s] = fmaf(-r0, sa, i0 * ca);
  }
}

__global__ void __launch_bounds__(THREADS)
quantum_layer_kernel(const float* __restrict__ x, const float* __restrict__ params,
                     float* __restrict__ out, int bsz) {
  __shared__ float sre[DIM], sim[DIM];        // state planes
  __shared__ float ec[DIM], es[DIM];          // encoding phase cos/sin
  __shared__ float Msh[256];                  // 16x16 group rotation matrix
  __shared__ float psh[NBLOCKS * 6 * N_WIRES];
  __shared__ float xsh[N_WIRES];
  __shared__ unsigned short permsh[DIM];
  __shared__ float oacc[N_WIRES];

  const int tid = threadIdx.x;
  const int b   = blockIdx.x;
  if (b >= bsz) return;

  if (tid < NBLOCKS * 6 * N_WIRES) psh[tid] = params[tid];
  if (tid < N_WIRES) {
    xsh[tid]  = x[b * N_WIRES + tid];
    oacc[tid] = 0.f;
  }
  __syncthreads();

  // init |0...0>, encoding phase table, permutation table
  #pragma unroll
  for (int r = 0; r < SPT; ++r) {
    int s = tid + r * THREADS;
    permsh[s] = (unsigned short)perm_index(s);
    sre[s] = (s == 0) ? 1.f : 0.f;
    sim[s] = 0.f;
    float a = 0.f;
    #pragma unroll
    for (int w = 0; w < N_WIRES; ++w) {
      float v = xsh[w];
      a += ((s >> (9 - w)) & 1) ? -v : v;
    }
    a *= 0.5f;
    fast_sincos(a, &es[s], &ec[s]);
  }
  __syncthreads();

  for (int k = 0; k < NBLOCKS; ++k) {
    const float* pb = psh + k * 6 * N_WIRES;
    rz_stage(sre, sim, pb + 0 * N_WIRES, tid);
    ry_all(sre, sim, Msh, pb + 1 * N_WIRES, tid);
    rz_stage(sre, sim, pb + 2 * N_WIRES, tid);
    apply_perm(sre, sim, permsh, tid);
    rz_stage(sre, sim, pb + 3 * N_WIRES, tid);
    ry_all(sre, sim, Msh, pb + 4 * N_WIRES, tid);
    rz_stage(sre, sim, pb + 5 * N_WIRES, tid);
    apply_perm(sre, sim, permsh, tid);
    if (k != NBLOCKS - 1) apply_enc(sre, sim, ec, es, tid);
  }
  __syncthreads();

  // out[b][w] = sum_s |psi_s|^2 * zval[w][s]
  float acc[N_WIRES];
  #pragma unroll
  for (int w = 0; w < N_WIRES; ++w) acc[w] = 0.f;
  #pragma unroll
  for (int r = 0; r < SPT; ++r) {
    int s = tid + r * THREADS;
    float pr = fmaf(sre[s], sre[s], sim[s] * sim[s]);
    #pragma unroll
    for (int w = 0; w < N_WIRES; ++w)
      acc[w] += ((s >> (9 - w)) & 1) ? -pr : pr;
  }
  #pragma unroll
  for (int w = 0; w < N_WIRES; ++w) atomicAdd(&oacc[w], acc[w]);
  __syncthreads();
  if (tid < N_WIRES) out[b * N_WIRES + tid] = oacc[tid];
}

extern "C" void kernel_launch(void* const* d_in, const int* in_sizes, int n_in,
                              void* d_out, int out_size, void* d_ws, size_t ws_size,
                              hipStream_t stream) {
  const float* x      = (const float*)d_in[0];   // (2048, 10) fp32
  const float* params = (const float*)d_in[1];   // (4, 6, 10) fp32
  float* out = (float*)d_out;                    // (1, 2048, 10) fp32
  (void)d_ws; (void)ws_size; (void)n_in; (void)out_size;
  int bsz = in_sizes[0] / N_WIRES;
  quantum_layer_kernel<<<dim3(bsz), dim3(THREADS), 0, stream>>>(x, params, out, bsz);
}